// AngularContrastiveLoss_86543591015184
// MI455X (gfx1250) — compile-verified
//
#include <hip/hip_runtime.h>
#include <math.h>

typedef __attribute__((ext_vector_type(2))) float v2f;
typedef __attribute__((ext_vector_type(8))) float v8f;

#define EPSV    1e-7f
#define MARGIN  0.5f
#define ALPHA   0.5f
#define INV_T   (1.0f / 0.06f)

#define RT_SCL  4     // row tiles per wave, SCL pass (K=128)
#define RT_AMC  2     // row tiles per wave, AMC pass (K=256)
#define CCH     16    // column chunks (gridDim.y) per Gram pass

// ---------------------------------------------------------------------------
__device__ __forceinline__ float reduce_low16(float v) {
    v += __shfl_xor(v, 1, 32);
    v += __shfl_xor(v, 2, 32);
    v += __shfl_xor(v, 4, 32);
    v += __shfl_xor(v, 8, 32);
    return v;
}

__global__ void k_zero(float* __restrict__ p, int n) {
    const int i = blockIdx.x * blockDim.x + threadIdx.x;
    if (i < n) p[i] = 0.0f;
}

// rows of length blockDim.x (<=256), one block per row: out = in / ||in||
__global__ void k_normalize_rows(const float* __restrict__ in,
                                 float* __restrict__ out, int cols) {
    __shared__ float red[256];
    const int row = blockIdx.x;
    const float v = in[(size_t)row * cols + threadIdx.x];
    red[threadIdx.x] = v * v;
    __syncthreads();
    for (int s = blockDim.x >> 1; s > 0; s >>= 1) {
        if (threadIdx.x < (unsigned)s) red[threadIdx.x] += red[threadIdx.x + s];
        __syncthreads();
    }
    out[(size_t)row * cols + threadIdx.x] = v / sqrtf(red[0]);
}

// normalize p1/p2 rows and write concatenated cf[2N][cols]
__global__ void k_normalize_cat(const float* __restrict__ p1,
                                const float* __restrict__ p2,
                                float* __restrict__ cf, int N, int cols) {
    __shared__ float red[256];
    const int row = blockIdx.x;
    const float* src = (row < N) ? (p1 + (size_t)row * cols)
                                 : (p2 + (size_t)(row - N) * cols);
    const float v = src[threadIdx.x];
    red[threadIdx.x] = v * v;
    __syncthreads();
    for (int s = blockDim.x >> 1; s > 0; s >>= 1) {
        if (threadIdx.x < (unsigned)s) red[threadIdx.x] += red[threadIdx.x + s];
        __syncthreads();
    }
    cf[(size_t)row * cols + threadIdx.x] = v / sqrtf(red[0]);
}

// ---------------------------------------------------------------------------
// SCL fused Gram pass, register-blocked: each wave owns RT_SCL 16-row bands
// and a chunk of column tiles. One B-fragment load feeds RT_SCL WMMA chains.
// Accumulates S[i] = sum_{j!=i} exp((sim-1)/T), Tm[i] = sum_{j!=i,same} (sim-1)/T.
// ---------------------------------------------------------------------------
__global__ void __launch_bounds__(32)
k_scl_gram(const float* __restrict__ cf, const int* __restrict__ labels,
           float* __restrict__ S, float* __restrict__ Tm,
           int N, int nColTiles) {
    constexpr int KS = 32;            // K = 128, 4 per wmma step
    constexpr int RT = RT_SCL;
    const int lane = threadIdx.x;
    const int m    = lane & 15;
    const int koff = (lane >> 4) * 2;
    const int rg   = blockIdx.x;                    // group of RT row tiles
    const int tpc  = nColTiles / gridDim.y;         // tiles per column chunk
    const int ct0  = blockIdx.y * tpc;

    // preload RT A fragments (persist in VGPRs for the whole column sweep)
    v2f a[RT][KS];
    int rbase[RT];
    int labI[RT][8];
#pragma unroll
    for (int t = 0; t < RT; ++t) {
        const int rtile = rg * RT + t;
        const float* ap = cf + (size_t)(rtile * 16 + m) * 128 + koff;
#pragma unroll
        for (int k = 0; k < KS; ++k) a[t][k] = *(const v2f*)(ap + k * 4);
        rbase[t] = rtile * 16 + ((lane >> 4) << 3);
#pragma unroll
        for (int r = 0; r < 8; ++r) {
            const int i = rbase[t] + r;
            labI[t][r] = (i < N) ? labels[i] : labels[i - N];
        }
    }

    float accS[RT][8], accT[RT][8];
#pragma unroll
    for (int t = 0; t < RT; ++t)
#pragma unroll
        for (int r = 0; r < 8; ++r) { accS[t][r] = 0.0f; accT[t][r] = 0.0f; }

    for (int ct = ct0; ct < ct0 + tpc; ++ct) {
        const float* bp = cf + (size_t)(ct * 16 + m) * 128 + koff;
        v8f c[RT];
#pragma unroll
        for (int t = 0; t < RT; ++t) c[t] = (v8f){};
#pragma unroll
        for (int k = 0; k < KS; ++k) {
            const v2f b = *(const v2f*)(bp + k * 4);   // 1 load ...
#pragma unroll
            for (int t = 0; t < RT; ++t)               // ... RT independent WMMAs
                c[t] = __builtin_amdgcn_wmma_f32_16x16x4_f32(
                           false, a[t][k], false, b, (short)0, c[t], false, false);
        }
        const int j    = ct * 16 + m;
        const int labJ = (j < N) ? labels[j] : labels[j - N];
#pragma unroll
        for (int t = 0; t < RT; ++t) {
#pragma unroll
            for (int r = 0; r < 8; ++r) {
                const int i = rbase[t] + r;
                if (i != j) {
                    const float logit = (c[t][r] - 1.0f) * INV_T; // row max == 1/T
                    accS[t][r] += __expf(logit);
                    if (labI[t][r] == labJ) accT[t][r] += logit;
                }
            }
        }
    }

#pragma unroll
    for (int t = 0; t < RT; ++t) {
#pragma unroll
        for (int r = 0; r < 8; ++r) {
            const float s = reduce_low16(accS[t][r]);
            const float u = reduce_low16(accT[t][r]);
            if (m == 0) {                       // lanes 0 and 16: each owns 8 rows
                atomicAdd(&S[rbase[t] + r], s);
                atomicAdd(&Tm[rbase[t] + r], u);
            }
        }
    }
}

// ---------------------------------------------------------------------------
// AMC fused Gram pass, register-blocked (RT_AMC row bands per wave, K=256).
// ---------------------------------------------------------------------------
__global__ void __launch_bounds__(32)
k_amc_gram(const float* __restrict__ f, const int* __restrict__ labels,
           float* __restrict__ sums, int N, int nColTiles) {
    constexpr int KS = 64;            // K = 256
    constexpr int RT = RT_AMC;
    const int lane = threadIdx.x;
    const int m    = lane & 15;
    const int koff = (lane >> 4) * 2;
    const int rg   = blockIdx.x;
    const int tpc  = nColTiles / gridDim.y;
    const int ct0  = blockIdx.y * tpc;

    v2f a[RT][KS];
    int rbase[RT];
    int labI[RT][8];
#pragma unroll
    for (int t = 0; t < RT; ++t) {
        const int rtile = rg * RT + t;
        const float* ap = f + (size_t)(rtile * 16 + m) * 256 + koff;
#pragma unroll
        for (int k = 0; k < KS; ++k) a[t][k] = *(const v2f*)(ap + k * 4);
        rbase[t] = rtile * 16 + ((lane >> 4) << 3);
#pragma unroll
        for (int r = 0; r < 8; ++r) labI[t][r] = labels[rbase[t] + r];
    }

    float acc = 0.0f;
    for (int ct = ct0; ct < ct0 + tpc; ++ct) {
        const float* bp = f + (size_t)(ct * 16 + m) * 256 + koff;
        v8f c[RT];
#pragma unroll
        for (int t = 0; t < RT; ++t) c[t] = (v8f){};
#pragma unroll
        for (int k = 0; k < KS; ++k) {
            const v2f b = *(const v2f*)(bp + k * 4);
#pragma unroll
            for (int t = 0; t < RT; ++t)
                c[t] = __builtin_amdgcn_wmma_f32_16x16x4_f32(
                           false, a[t][k], false, b, (short)0, c[t], false, false);
        }
        const int j    = ct * 16 + m;
        const int labJ = labels[j];
#pragma unroll
        for (int t = 0; t < RT; ++t) {
#pragma unroll
            for (int r = 0; r < 8; ++r) {
                const int i = rbase[t] + r;
                if (i != j) {
                    const float sim = fminf(fmaxf(c[t][r], -1.0f + EPSV), 1.0f - EPSV);
                    const float ang = acosf(sim);
                    if (labI[t][r] == labJ) {
                        acc += ang * ang;             // positive pair
                    } else {
                        const float d = MARGIN - ang; // negative pair
                        if (d > 0.0f) acc += d * d;
                    }
                }
            }
        }
    }
    acc += __shfl_xor(acc, 1, 32);
    acc += __shfl_xor(acc, 2, 32);
    acc += __shfl_xor(acc, 4, 32);
    acc += __shfl_xor(acc, 8, 32);
    acc += __shfl_xor(acc, 16, 32);
    if (lane == 0) atomicAdd(&sums[0], acc);
}

// ---------------------------------------------------------------------------
__global__ void k_scl_finalize(const float* __restrict__ S,
                               const float* __restrict__ Tm,
                               const int* __restrict__ labels,
                               float* __restrict__ sums, int N, int twoN) {
    __shared__ float red[256];
    const int i = blockIdx.x * blockDim.x + threadIdx.x;
    float val = 0.0f;
    if (i < twoN) {
        const int li = (i < N) ? labels[i] : labels[i - N];
        int cnt = 0;
        for (int k = 0; k < N; ++k) cnt += (labels[k] == li) ? 1 : 0;
        const float P = (float)(2 * cnt - 1);           // positives per row
        val = -(Tm[i] / P - logf(S[i]));
    }
    red[threadIdx.x] = val;
    __syncthreads();
    for (int s = blockDim.x >> 1; s > 0; s >>= 1) {
        if (threadIdx.x < (unsigned)s) red[threadIdx.x] += red[threadIdx.x + s];
        __syncthreads();
    }
    if (threadIdx.x == 0) atomicAdd(&sums[1], red[0]);
}

__global__ void k_combine(const float* __restrict__ sums,
                          float* __restrict__ out, int twoN) {
    if (threadIdx.x == 0) {
        const float loss1 = sums[1] / (float)twoN;  // mean over 2N rows
        const float loss2 = sums[0] / 50.0f;        // (l1+l2)/50
        out[0] = ALPHA * loss1 + (1.0f - ALPHA) * loss2;
    }
}

// ---------------------------------------------------------------------------
extern "C" void kernel_launch(void* const* d_in, const int* in_sizes, int n_in,
                              void* d_out, int out_size, void* d_ws, size_t ws_size,
                              hipStream_t stream) {
    const float* am     = (const float*)d_in[0];   // [N, 256]
    const float* p1     = (const float*)d_in[1];   // [N, 128]
    const float* p2     = (const float*)d_in[2];   // [N, 128]
    const int*   labels = (const int*)d_in[3];     // [N]

    const int N    = in_sizes[3];        // 4096
    const int D    = in_sizes[0] / N;    // 256
    const int P    = in_sizes[1] / N;    // 128
    const int twoN = 2 * N;

    float* ws    = (float*)d_ws;
    float* fnorm = ws;                              // N*D
    float* cf    = fnorm + (size_t)N * D;           // 2N*P
    float* S     = cf + (size_t)twoN * P;           // 2N  (atomic-accumulated)
    float* Tm    = S + twoN;                        // 2N  (atomic-accumulated)
    float* sums  = Tm + twoN;                       // [amc_sum, scl_sum]
    float* out   = (float*)d_out;

    const int nz = 2 * twoN + 2;                    // S, Tm, sums are contiguous
    k_zero<<<(nz + 255) / 256, 256, 0, stream>>>(S, nz);
    k_normalize_rows<<<N, D, 0, stream>>>(am, fnorm, D);
    k_normalize_cat<<<twoN, P, 0, stream>>>(p1, p2, cf, N, P);

    dim3 gscl(twoN / 16 / RT_SCL, CCH);             // 128 x 16 waves
    k_scl_gram<<<gscl, 32, 0, stream>>>(cf, labels, S, Tm, N, twoN / 16);

    dim3 gamc(N / 16 / RT_AMC, CCH);                // 128 x 16 waves
    k_amc_gram<<<gamc, 32, 0, stream>>>(fnorm, labels, sums, N, N / 16);

    k_scl_finalize<<<(twoN + 255) / 256, 256, 0, stream>>>(S, Tm, labels, sums, N, twoN);
    k_combine<<<1, 32, 0, stream>>>(sums, out, twoN);
}